// PeriodicKNN_PyG_22179211117016
// MI455X (gfx1250) — compile-verified
//
#include <hip/hip_runtime.h>
#include <stdint.h>

// Periodic (minimum-image) k-NN, N=8192, k=17, MI455X / gfx1250.
//
// Strategy (compute-bound, selection-heavy — see analysis):
//  * Stage all 8192x3 f32 positions (96 KB) into LDS per workgroup using the
//    CDNA5 async global->LDS DMA path (global_load_async_to_lds_b128,
//    s_wait_asynccnt) — exploits the 320 KB WGP LDS.
//  * Block = 256 threads = 8 wave32 waves. Wave w scans candidate slice w
//    (1024 candidates) for 32 queries (one per lane): all lanes of a wave read
//    the SAME candidate -> conflict-free LDS broadcast; 3x ds_load_b128 per 4
//    candidates (AoS stride 12B, 16B-aligned groups of 4).
//  * Per-thread register-resident sorted top-17 via fully unrolled
//    compare/select insertion (no dynamic register indexing -> no spills).
//  * Partials packed as (f32bits(d2)<<32 | idx) u64 in LDS; wave 0 merges the
//    8 sorted 17-lists per query by repeated min-extraction (float bits of
//    nonnegative d2 are order-isomorphic to u32; ties break to lower index).

#define NPTS      8192
#define KNN       17
#define QPB       32                 // queries per block (== wave32 width)
#define SLICES    8                  // candidate slices (== waves per block)
#define TPB       (QPB * SLICES)     // 256 threads
#define SLICE_LEN (NPTS / SLICES)    // 1024 candidates per slice

__global__ __launch_bounds__(TPB)
void knn_periodic_kernel(const float* __restrict__ pos,
                         const float* __restrict__ box,
                         float* __restrict__ out) {
    __shared__ float              sPos[NPTS * 3];                 // 98304 B
    __shared__ unsigned long long sMerge[QPB][SLICES * KNN];      // 34816 B

    const int tid    = threadIdx.x;
    const int qLocal = tid & (QPB - 1);   // lane within wave = query
    const int slice  = tid >> 5;          // wave id = candidate slice
    const int qi     = blockIdx.x * QPB + qLocal;

    // ---- Stage all positions into LDS via CDNA5 async global->LDS DMA ----
    {
        const int nChunks = (NPTS * 3 * 4) / 16;   // 6144 x b128
        for (int c = tid; c < nChunks; c += TPB) {
            // Low 32 bits of a flat shared pointer == wave-relative LDS byte
            // offset (aperture layout), which is what VDST of the async op
            // expects.
            unsigned    ldsOff = (unsigned)(uintptr_t)(&sPos[c * 4]);
            const void* gaddr  = (const void*)((const char*)pos + (size_t)c * 16);
            asm volatile("global_load_async_to_lds_b128 %0, %1, off"
                         :: "v"(ldsOff), "v"(gaddr)
                         : "memory");
        }
        asm volatile("s_wait_asynccnt 0" ::: "memory");
    }
    __syncthreads();

    const float Lx = box[0], Ly = box[1], Lz = box[2];
    const float iLx = 1.0f / Lx, iLy = 1.0f / Ly, iLz = 1.0f / Lz;

    const float xq = sPos[qi * 3 + 0];
    const float yq = sPos[qi * 3 + 1];
    const float zq = sPos[qi * 3 + 2];

    // Register-resident sorted (ascending) top-K.
    float bd[KNN];
    int   bi[KNN];
#pragma unroll
    for (int t = 0; t < KNN; ++t) { bd[t] = 3.4e38f; bi[t] = 0; }

    // Fully-unrolled sorted insertion: compare/select bubble pass, executed
    // only when the candidate beats the current worst (wave32 keeps the
    // any-lane-diverges probability low).
    auto consider = [&](float px, float py, float pz, int j) {
        float dx = xq - px; dx -= Lx * rintf(dx * iLx);
        float dy = yq - py; dy -= Ly * rintf(dy * iLy);
        float dz = zq - pz; dz -= Lz * rintf(dz * iLz);
        float d2 = fmaf(dx, dx, fmaf(dy, dy, dz * dz));
        if (j != qi && d2 < bd[KNN - 1]) {
            float cd = d2; int ci = j;
#pragma unroll
            for (int s = 0; s < KNN; ++s) {
                bool  sw = cd < bd[s];
                float td = bd[s]; int ti = bi[s];
                bd[s] = sw ? cd : td;  bi[s] = sw ? ci : ti;
                cd    = sw ? td : cd;  ci    = sw ? ti : ci;
            }
        }
    };

    // ---- Scan this wave's candidate slice: 3x ds_load_b128 per 4 points ----
    const int j0 = slice * SLICE_LEN;
    const float4* p4 = (const float4*)sPos;
#pragma unroll 2
    for (int m = 0; m < SLICE_LEN / 4; ++m) {
        int j  = j0 + 4 * m;
        int f4 = (3 * j) >> 2;           // 12B * j is 16B-aligned (j % 4 == 0)
        float4 c0 = p4[f4 + 0];
        float4 c1 = p4[f4 + 1];
        float4 c2 = p4[f4 + 2];
        consider(c0.x, c0.y, c0.z, j + 0);
        consider(c0.w, c1.x, c1.y, j + 1);
        consider(c1.z, c1.w, c2.x, j + 2);
        consider(c2.y, c2.z, c2.w, j + 3);
    }

    // ---- Publish per-(query,slice) partial top-K as packed u64 ----
#pragma unroll
    for (int t = 0; t < KNN; ++t) {
        unsigned long long pk =
            ((unsigned long long)__float_as_uint(bd[t]) << 32) | (unsigned)bi[t];
        sMerge[qLocal][slice * KNN + t] = pk;
    }
    __syncthreads();

    // ---- Wave 0 merges SLICES sorted lists per query (min-extraction) ----
    if (tid < QPB) {
        const int q  = tid;
        const int gq = blockIdx.x * QPB + q;
        const int NK = NPTS * KNN;
        float* outSrc  = out;             // edge_index[0] : source ids
        float* outIdx  = out + NK;        // edge_index[1] : neighbor ids
        float* outDist = out + 2 * NK;    // knn_dist
        for (int t = 0; t < KNN; ++t) {
            unsigned long long best = 0xFFFFFFFFFFFFFFFFull;
            int bp = 0;
            for (int e = 0; e < SLICES * KNN; ++e) {
                unsigned long long v = sMerge[q][e];
                if (v < best) { best = v; bp = e; }
            }
            sMerge[q][bp] = 0xFFFFFFFFFFFFFFFFull;   // consume entry
            float d2 = __uint_as_float((unsigned)(best >> 32));
            int   jj = (int)(best & 0xFFFFFFFFu);
            outSrc [gq * KNN + t] = (float)gq;
            outIdx [gq * KNN + t] = (float)jj;
            outDist[gq * KNN + t] = sqrtf(d2);
        }
    }
}

extern "C" void kernel_launch(void* const* d_in, const int* in_sizes, int n_in,
                              void* d_out, int out_size, void* d_ws, size_t ws_size,
                              hipStream_t stream) {
    (void)in_sizes; (void)n_in; (void)d_ws; (void)ws_size; (void)out_size;
    const float* pos = (const float*)d_in[0];   // [N,3] f32
    const float* box = (const float*)d_in[1];   // [3]  f32
    float*       out = (float*)d_out;           // [src | idx | dist], 3*N*k f32

    dim3 grid(NPTS / QPB);   // 256 blocks
    dim3 block(TPB);         // 256 threads (8 wave32 waves)
    knn_periodic_kernel<<<grid, block, 0, stream>>>(pos, box, out);
}